// SparseNetwork_2241972928735
// MI455X (gfx1250) — compile-verified
//
#include <hip/hip_runtime.h>

typedef __attribute__((ext_vector_type(2))) float v2f;
typedef __attribute__((ext_vector_type(8))) float v8f;

#define NTHREADS 1024

// LDS accumulator layout (64 floats):
//   [ 0: 8)  A0 partials  : sum of first-layer net scalars (grouping irrelevant,
//                           only the total is needed; 8 slots spread contention)
//   [ 8:40)  AH partials  : AH[l] total in slots [8+8l, 8+8l+8)
//   [40:56)  AL            : last-layer group sums (o = m/8, 10 used)
//   [56:64)  S             : batch sums of x
__global__ __launch_bounds__(NTHREADS)
void sparse_mlp_fused(const float* __restrict__ x,
                      const float* __restrict__ fw1, const float* __restrict__ fw2,
                      const float* __restrict__ fw3,
                      const float* __restrict__ hw1, const float* __restrict__ hw2,
                      const float* __restrict__ hw3,
                      const float* __restrict__ lw1, const float* __restrict__ lw2,
                      const float* __restrict__ lw3,
                      float* __restrict__ out)
{
    __shared__ float acc[64];
    __shared__ float srow[8];

    const int tid = threadIdx.x;
    if (tid < 64) acc[tid] = 0.0f;
    __syncthreads();

    // ---- Phase 1a: first-layer net scalars a_n; only the grand total matters.
    // 6272 nets = 896 threads x 7 nets; one LDS atomic per thread.
    if (tid < 896) {
        const float4* w2v = (const float4*)fw2;
        const float2* w3v = (const float2*)fw3;
        float local = 0.0f;
        int n = tid * 7;
#pragma unroll
        for (int k = 0; k < 7; ++k, ++n) {
            float  w0 = fw1[n * 10 + 4];
            float  w1 = fw1[n * 10 + 9];
            float4 w2 = w2v[n];                 // [g0h0, g0h1, g1h0, g1h1]
            float2 w3 = w3v[n];
            float  u0 = w2.x * w0 + w2.y * w1;
            float  u1 = w2.z * w0 + w2.w * w1;
            local += w3.x * u0 + w3.y * u1;
        }
        atomicAdd(&acc[tid & 7], local);
    }

    // ---- Phase 1b: batch sums S_b. 6272 = 784 threads x 8 elems; 784 % 8 == 0
    // so each chunk stays inside one batch row: b = tid / 98.
    if (tid < 784) {
        const float4* xv = (const float4*)x;
        float4 p0 = xv[tid * 2 + 0];
        float4 p1 = xv[tid * 2 + 1];
        float local = ((p0.x + p0.y) + (p0.z + p0.w)) +
                      ((p1.x + p1.y) + (p1.z + p1.w));
        atomicAdd(&acc[56 + tid / 98], local);
    }

    // ---- Phase 1c: hidden nets (4 layers x 64); only per-layer totals matter.
    if (tid < 256) {
        const float4* w2v = (const float4*)hw2;
        const float2* w3v = (const float2*)hw3;
        int m = tid;                            // flat net id = l*64 + j
        float  w0 = hw1[m * 10 + 4];
        float  w1 = hw1[m * 10 + 9];
        float4 w2 = w2v[m];
        float2 w3 = w3v[m];
        float  u0 = w2.x * w0 + w2.y * w1;
        float  u1 = w2.z * w0 + w2.w * w1;
        float  an = w3.x * u0 + w3.y * u1;
        int l = m >> 6;
        atomicAdd(&acc[8 + l * 8 + (m & 7)], an);
    }

    // ---- Phase 1d: last-layer nets (80); group sums AL[o], o = m/8.
    if (tid < 80) {
        const float4* w2v = (const float4*)lw2;
        const float2* w3v = (const float2*)lw3;
        int m = tid;
        float  w0 = lw1[m * 10 + 4];
        float  w1 = lw1[m * 10 + 9];
        float4 w2 = w2v[m];
        float2 w3 = w3v[m];
        float  u0 = w2.x * w0 + w2.y * w1;
        float  u1 = w2.z * w0 + w2.w * w1;
        float  an = w3.x * u0 + w3.y * u1;
        atomicAdd(&acc[40 + (m >> 3)], an);
    }

    __syncthreads();

    // ---- Phase 2 (wave 0): exact collapse.
    // Row sums propagate as R <- R*(1 + sum(AH_l)), so
    //   out[b,o] = S_b * sum(A0) * prod_l(1 + sum(AH_l)) * AL_o
    // One WMMA f32 16x16x4 produces the 8x10 outer product in C/D layout.
    if (tid < 32) {
        const int lane = tid;

        // lanes 0..4 each total one 8-slot group: 0 -> A0, 1..4 -> AH[l-1]
        if (lane < 5) {
            float s = 0.0f;
#pragma unroll
            for (int i = 0; i < 8; ++i) s += acc[8 * lane + i];
            srow[lane] = s;
        }
        // all lanes read back (same-wave LDS ordering; compiler inserts waits)
        float coef = srow[0] * (1.0f + srow[1]) * (1.0f + srow[2]) *
                               (1.0f + srow[3]) * (1.0f + srow[4]);

        // A (16x4): column K=0 holds S_b * coef (rows b<8); B (4x16): row K=0
        // holds AL_o (cols o<10). D = A x B + 0.
        v2f a, b;
        a[0] = (lane < 8) ? acc[56 + lane] * coef : 0.0f;
        a[1] = 0.0f;
        int col = lane & 15;
        b[0] = (lane < 16 && col < 10) ? acc[40 + col] : 0.0f;
        b[1] = 0.0f;
        v8f z = {0.0f, 0.0f, 0.0f, 0.0f, 0.0f, 0.0f, 0.0f, 0.0f};
        v8f d = __builtin_amdgcn_wmma_f32_16x16x4_f32(
                    false, a, false, b, (short)0, z, false, false);

        // store [8,10] row-major: element (M=r, N=lane), lanes 0..9
        if (lane < 10) {
#pragma unroll
            for (int r = 0; r < 8; ++r) out[r * 10 + lane] = d[r];
        }
    }
}

extern "C" void kernel_launch(void* const* d_in, const int* in_sizes, int n_in,
                              void* d_out, int out_size, void* d_ws, size_t ws_size,
                              hipStream_t stream) {
    (void)in_sizes; (void)n_in; (void)out_size; (void)d_ws; (void)ws_size;
    const float* x   = (const float*)d_in[0];
    const float* fw1 = (const float*)d_in[1];
    const float* fw2 = (const float*)d_in[2];
    const float* fw3 = (const float*)d_in[3];
    const float* hw1 = (const float*)d_in[4];
    const float* hw2 = (const float*)d_in[5];
    const float* hw3 = (const float*)d_in[6];
    const float* lw1 = (const float*)d_in[7];
    const float* lw2 = (const float*)d_in[8];
    const float* lw3 = (const float*)d_in[9];

    sparse_mlp_fused<<<1, NTHREADS, 0, stream>>>(
        x, fw1, fw2, fw3, hw1, hw2, hw3, lw1, lw2, lw3, (float*)d_out);
}